// SkipGNN_72060961292400
// MI455X (gfx1250) — compile-verified
//
#include <hip/hip_runtime.h>

typedef __attribute__((ext_vector_type(2))) float v2f;
typedef __attribute__((ext_vector_type(8))) float v8f;

#define NNODES 100000
#define D 128  // D_IN == D_OUT == 128

// ---------------------------------------------------------------------------
// Kernel 1: fused dual GEMM with FP32 WMMA (16x16x4).
//   sup = x @ W                (workspace)
//   out = x @ W_skip + b       (pre-initializes d_out; SpMM adds on top)
// Block = 256 threads = 8 waves; block owns a 16-row M tile, wave w owns
// N-tile w. x tile staged in LDS. N = 6250*16 exactly -> EXEC all-1s.
// ---------------------------------------------------------------------------
__global__ __launch_bounds__(256) void gemm_dual_f32(
    const float* __restrict__ x, const float* __restrict__ W,
    const float* __restrict__ Wskip, const float* __restrict__ bias,
    float* __restrict__ sup, float* __restrict__ out)
{
    __shared__ float xs[16 * D];  // 8 KB

    const int tid    = threadIdx.x;
    const int m_base = blockIdx.x * 16;

    const float4* xg = (const float4*)(x + (size_t)m_base * D);
    float4*       xl = (float4*)xs;
    for (int i = tid; i < (16 * D) / 4; i += 256) xl[i] = xg[i];
    __syncthreads();

    const int wave   = tid >> 5;
    const int lane   = tid & 31;
    const int n_base = wave * 16;
    const int half   = lane >> 4;   // 0: lanes 0-15, 1: lanes 16-31
    const int ml     = lane & 15;   // M (A) / N (B,C,D) within tile

    v8f acc1 = {};
    v8f acc2 = {};

    for (int k = 0; k < D; k += 4) {
        const int k0 = k + half * 2;   // A-layout: half selects K pair
        v2f a;
        a.x = xs[ml * D + k0 + 0];
        a.y = xs[ml * D + k0 + 1];
        v2f b1, b2;
        b1.x = W[(size_t)(k0 + 0) * D + n_base + ml];
        b1.y = W[(size_t)(k0 + 1) * D + n_base + ml];
        b2.x = Wskip[(size_t)(k0 + 0) * D + n_base + ml];
        b2.y = Wskip[(size_t)(k0 + 1) * D + n_base + ml];
        acc1 = __builtin_amdgcn_wmma_f32_16x16x4_f32(false, a, false, b1,
                                                     (short)0, acc1, false, false);
        acc2 = __builtin_amdgcn_wmma_f32_16x16x4_f32(false, a, false, b2,
                                                     (short)0, acc2, false, false);
    }

    const float bv = bias[n_base + ml];
#pragma unroll
    for (int r = 0; r < 8; ++r) {
        const int row = m_base + r + half * 8;
        const int col = n_base + ml;
        sup[(size_t)row * D + col] = acc1[r];
        out[(size_t)row * D + col] = acc2[r] + bv;
    }
}

// ---------------------------------------------------------------------------
// CSR build step 1: zero per-row counts.
// ---------------------------------------------------------------------------
__global__ __launch_bounds__(256) void zero_counts(int* __restrict__ counts) {
    const int i = blockIdx.x * blockDim.x + threadIdx.x;
    if (i < NNODES) counts[i] = 0;
}

// ---------------------------------------------------------------------------
// CSR build step 2: histogram of edges per destination row (int atomics).
// ---------------------------------------------------------------------------
__global__ __launch_bounds__(256) void count_edges(
    const int* __restrict__ rows, int* __restrict__ counts, int E)
{
    const int e = blockIdx.x * blockDim.x + threadIdx.x;
    if (e < E) atomicAdd(&counts[rows[e]], 1);
}

// ---------------------------------------------------------------------------
// CSR build step 3: exclusive scan over 100k counts. Single block of 1024
// threads: per-thread serial chunk sum, Hillis-Steele scan in LDS over the
// 1024 partials, then per-thread prefix write. Also seeds cursor = rowStart.
// ---------------------------------------------------------------------------
__global__ __launch_bounds__(1024) void scan_rows(
    const int* __restrict__ counts, int* __restrict__ rowStart,
    int* __restrict__ cursor)
{
    __shared__ int partial[1024];
    const int t     = threadIdx.x;
    const int chunk = (NNODES + 1023) / 1024;  // 98
    const int beg   = t * chunk;
    const int end   = (beg + chunk < NNODES) ? beg + chunk : NNODES;

    int s = 0;
    for (int i = beg; i < end; ++i) s += counts[i];
    partial[t] = s;
    __syncthreads();

    for (int off = 1; off < 1024; off <<= 1) {   // inclusive scan
        int v = 0;
        if (t >= off) v = partial[t - off];
        __syncthreads();
        if (t >= off) partial[t] += v;
        __syncthreads();
    }

    int run = (t == 0) ? 0 : partial[t - 1];     // exclusive base
    for (int i = beg; i < end; ++i) {
        rowStart[i] = run;
        cursor[i]   = run;
        run += counts[i];
    }
    if (t == 1023) rowStart[NNODES] = run;       // == E (tail chunks empty)
}

// ---------------------------------------------------------------------------
// CSR build step 4: scatter edge payloads into row-sorted order.
// ---------------------------------------------------------------------------
__global__ __launch_bounds__(256) void scatter_edges(
    const int* __restrict__ rows, const int* __restrict__ cols,
    const float* __restrict__ ew, int* __restrict__ cursor,
    int* __restrict__ eCol, float* __restrict__ eW, int E)
{
    const int e = blockIdx.x * blockDim.x + threadIdx.x;
    if (e >= E) return;
    const int pos = atomicAdd(&cursor[rows[e]], 1);
    eCol[pos] = cols[e];
    eW[pos]   = ew[e];
}

// ---------------------------------------------------------------------------
// SpMM: one wave per output row, zero FP atomics. Each lane owns a float4
// slice of the 128-wide row; loop over the row's CSR segment gathering
// L2-resident support1 rows; single RMW of the output row at the end.
// 2-way unrolled with dual accumulators for memory-latency hiding.
// N = 100000 rows = 12500 blocks * 8 waves exactly.
// ---------------------------------------------------------------------------
__global__ __launch_bounds__(256) void row_reduce(
    const float* __restrict__ sup, const int* __restrict__ rowStart,
    const int* __restrict__ eCol, const float* __restrict__ eW,
    float* __restrict__ out)
{
    const int row  = (blockIdx.x * blockDim.x + threadIdx.x) >> 5;
    const int lane = threadIdx.x & 31;

    const int beg = rowStart[row];
    const int end = rowStart[row + 1];

    float4 acc0 = {0.f, 0.f, 0.f, 0.f};
    float4 acc1 = {0.f, 0.f, 0.f, 0.f};

    int j = beg;
    for (; j + 1 < end; j += 2) {
        const int   c0 = eCol[j];
        const float w0 = eW[j];
        const int   c1 = eCol[j + 1];
        const float w1 = eW[j + 1];
        const float4 v0 = *(const float4*)(sup + (size_t)c0 * D + lane * 4);
        const float4 v1 = *(const float4*)(sup + (size_t)c1 * D + lane * 4);
        acc0.x += w0 * v0.x; acc0.y += w0 * v0.y;
        acc0.z += w0 * v0.z; acc0.w += w0 * v0.w;
        acc1.x += w1 * v1.x; acc1.y += w1 * v1.y;
        acc1.z += w1 * v1.z; acc1.w += w1 * v1.w;
    }
    if (j < end) {
        const int   c = eCol[j];
        const float w = eW[j];
        const float4 v = *(const float4*)(sup + (size_t)c * D + lane * 4);
        acc0.x += w * v.x; acc0.y += w * v.y;
        acc0.z += w * v.z; acc0.w += w * v.w;
    }

    float4* o = (float4*)(out + (size_t)row * D + lane * 4);
    float4  r = *o;
    r.x += acc0.x + acc1.x;
    r.y += acc0.y + acc1.y;
    r.z += acc0.z + acc1.z;
    r.w += acc0.w + acc1.w;
    *o = r;
}

// ---------------------------------------------------------------------------
// Fallback SpMM (small workspace): one wave per edge, f32 atomic scatter.
// ---------------------------------------------------------------------------
__global__ __launch_bounds__(256) void spmm_scatter(
    const float* __restrict__ sup, const int* __restrict__ rows,
    const int* __restrict__ cols, const float* __restrict__ ew,
    float* __restrict__ out, int E)
{
    const int gid  = blockIdx.x * blockDim.x + threadIdx.x;
    const int e    = gid >> 5;
    const int lane = gid & 31;
    if (e >= E) return;
    const int   c = cols[e];
    const int   r = rows[e];
    const float w = ew[e];
    const float4 v = *(const float4*)(sup + (size_t)c * D + lane * 4);
    float* o = out + (size_t)r * D + lane * 4;
    atomicAdd(o + 0, w * v.x);
    atomicAdd(o + 1, w * v.y);
    atomicAdd(o + 2, w * v.z);
    atomicAdd(o + 3, w * v.w);
}

// ---------------------------------------------------------------------------
// Inputs: x[N*128], rows[E], cols[E], edge_w[E], W[128^2], W_skip[128^2], b[128]
// Workspace layout (bytes, 512-aligned slabs):
//   sup      : N*128*4          = 51,200,000
//   counts   : N*4              =    400,384 (padded)
//   rowStart : (N+1)*4          =    400,384 (padded)
//   cursor   : N*4              =    400,384 (padded)
//   eCol     : E*4              = 12,800,000
//   eW       : E*4              = 12,800,000   -> total ~78.0 MB
// ---------------------------------------------------------------------------
extern "C" void kernel_launch(void* const* d_in, const int* in_sizes, int n_in,
                              void* d_out, int out_size, void* d_ws, size_t ws_size,
                              hipStream_t stream) {
    const float* x     = (const float*)d_in[0];
    const int*   rows  = (const int*)d_in[1];
    const int*   cols  = (const int*)d_in[2];
    const float* ew    = (const float*)d_in[3];
    const float* W     = (const float*)d_in[4];
    const float* Wskip = (const float*)d_in[5];
    const float* bias  = (const float*)d_in[6];
    float*       out   = (float*)d_out;

    const int E = in_sizes[1];

    char* ws = (char*)d_ws;
    const size_t supBytes   = (size_t)NNODES * D * 4;       // 51,200,000
    const size_t cntBytes   = ((size_t)NNODES * 4 + 511) & ~(size_t)511;
    const size_t rsBytes    = ((size_t)(NNODES + 1) * 4 + 511) & ~(size_t)511;
    const size_t curBytes   = cntBytes;
    const size_t eColBytes  = (size_t)E * 4;
    const size_t eWBytes    = (size_t)E * 4;

    size_t off = 0;
    float* sup      = (float*)(ws + off); off += supBytes;
    int*   counts   = (int*)  (ws + off); off += cntBytes;
    int*   rowStart = (int*)  (ws + off); off += rsBytes;
    int*   cursor   = (int*)  (ws + off); off += curBytes;
    int*   eCol     = (int*)  (ws + off); off += eColBytes;
    float* eWs      = (float*)(ws + off); off += eWBytes;
    const size_t needFull = off;

    // Dual GEMM + bias (initializes d_out so SpMM accumulates on top).
    gemm_dual_f32<<<NNODES / 16, 256, 0, stream>>>(x, W, Wskip, bias, sup, out);

    const int eBlocks = (E + 255) / 256;

    if (ws_size >= needFull) {
        // CSR path: no floating-point atomics.
        zero_counts<<<(NNODES + 255) / 256, 256, 0, stream>>>(counts);
        count_edges<<<eBlocks, 256, 0, stream>>>(rows, counts, E);
        scan_rows<<<1, 1024, 0, stream>>>(counts, rowStart, cursor);
        scatter_edges<<<eBlocks, 256, 0, stream>>>(rows, cols, ew, cursor,
                                                   eCol, eWs, E);
        row_reduce<<<(NNODES * 32) / 256, 256, 0, stream>>>(sup, rowStart,
                                                            eCol, eWs, out);
    } else {
        // Fallback: wave-per-edge atomic scatter.
        const long long threads = (long long)E * 32;
        spmm_scatter<<<(int)((threads + 255) / 256), 256, 0, stream>>>(
            sup, rows, cols, ew, out, E);
    }
}